// Lattice_23063974379522
// MI455X (gfx1250) — compile-verified
//
#include <hip/hip_runtime.h>
#include <hip/hip_bf16.h>

// Problem constants fixed by the reference's setup_inputs().
#define NEGV (-1.0e9f)
constexpr int S_   = 10;
constexpr int T_   = 1024;
constexpr int L_   = 16;
constexpr int K_   = 1024;       // 2^S
constexpr int TP1  = T_ + 1;     // 1025
constexpr int LP1  = L_ + 1;     // 17
constexpr int ROWS = T_ + 2;     // 1026 rows in A / B tables

typedef float v2f __attribute__((ext_vector_type(2)));
typedef float v8f __attribute__((ext_vector_type(8)));

// ---------------------------------------------------------------------------
// WMMA phase: compute row_c[j, k] = logsumexp_d( E[j][d] + W[d][k] ) for the 5
// row-slots over all 1024 states, as  M_j + N_k + log( exp(E-M) @ exp(W-N) )
// using V_WMMA_F32_16X16X4_F32 (16x20x16 tiles, 5 chained k-steps of 4).
//   W[d][k] = g[(row0 + step*d)*K + k]  for d <= maxD, else NEGV  (d<17)
// sEa: exp-normalized E, 16 rows x 20 cols (rows 5..15 and cols 17..19 = 0)
// ---------------------------------------------------------------------------
__device__ __forceinline__ void rowc_wmma(int tid,
                                          const float* __restrict__ g,
                                          int row0, int step, int maxD,
                                          const float* sEa, const float* sM,
                                          const float* sN, float* sRowC)
{
    const int lane = tid & 31;
    const int wave = tid >> 5;
    const int half = lane >> 4;   // 0 or 1
    const int col  = lane & 15;

    #pragma unroll
    for (int tt = 0; tt < 2; ++tt) {
        const int kbase = wave * 32 + tt * 16;
        const int ck    = kbase + col;       // this lane's state column
        const float nk  = sN[ck];

        v8f acc = {};  // zero C accumulator
        #pragma unroll
        for (int kk = 0; kk < 5; ++kk) {
            v2f a, b;
            #pragma unroll
            for (int v = 0; v < 2; ++v) {
                const int d = kk * 4 + half * 2 + v;
                // A-frag: lane holds slot-row M=col, K-index d.
                a[v] = sEa[col * 20 + d];
                // B-frag: lane holds state-col N=col, K-row d.
                float bv = 0.0f;
                if (d < LP1) {
                    float w = (d <= maxD) ? g[(size_t)(row0 + step * d) * K_ + ck]
                                          : NEGV;
                    bv = __expf(w - nk);
                }
                b[v] = bv;
            }
            acc = __builtin_amdgcn_wmma_f32_16x16x4_f32(
                      false, a, false, b, (short)0, acc, false, false);
        }
        // C layout: lane holds (M = r + 8*half, N = col). Keep slots 0..4.
        #pragma unroll
        for (int r = 0; r < 8; ++r) {
            const int slot = r + 8 * half;
            if (slot < 5) {
                float p = acc[r];
                sRowC[slot * K_ + ck] = sM[slot] + nk + __logf(p);
            }
        }
    }
}

// ---------------------------------------------------------------------------
// Forward scan (block 0): A[i+1][k] = LSE( A[i][k],
//    for set bit j<5:  row_c[j, k^2^j],   for set bit j>=5: tok[j,i]+A[i][k^2^j] )
// ---------------------------------------------------------------------------
__device__ void fwd_scan(const float* __restrict__ sb, const float* __restrict__ tokw,
                         float* __restrict__ A, float* __restrict__ out,
                         float* sEa, float* sM, float* sN, float* sRowC)
{
    const int k = threadIdx.x;
    A[k] = (k == 0) ? 0.0f : NEGV;
    __syncthreads();

    for (int i = 0; i <= T_; ++i) {
        const int maxD = (i < L_) ? i : L_;
        // Phase 0: normalized E rows + per-state window max N[k].
        if (k < 16) {
            float M = NEGV;
            if (k < 5) {
                float tmp[LP1];
                #pragma unroll
                for (int d = 0; d < LP1; ++d) {
                    float v = (d <= i) ? sb[(k * TP1 + (i - d)) * LP1 + d] : NEGV;
                    tmp[d] = v; M = fmaxf(M, v);
                }
                #pragma unroll
                for (int d = 0; d < LP1; ++d) sEa[k * 20 + d] = __expf(tmp[d] - M);
                sEa[k * 20 + 17] = 0.f; sEa[k * 20 + 18] = 0.f; sEa[k * 20 + 19] = 0.f;
            } else {
                #pragma unroll
                for (int d = 0; d < 20; ++d) sEa[k * 20 + d] = 0.f;
            }
            sM[k] = M;
        }
        {
            float m = NEGV;
            #pragma unroll
            for (int d = 0; d < LP1; ++d) {
                float v = (d <= maxD) ? A[(size_t)(i - d) * K_ + k] : NEGV;
                m = fmaxf(m, v);
            }
            sN[k] = m;
        }
        __syncthreads();

        rowc_wmma(k, A, i, -1, maxD, sEa, sM, sN, sRowC);
        __syncthreads();

        // Phase B: combine candidates with two-pass LSE.
        {
            const float prev = A[(size_t)i * K_ + k];
            float m = prev;
            #pragma unroll
            for (int j = 0; j < 5; ++j)
                if ((k >> j) & 1) m = fmaxf(m, sRowC[j * K_ + (k ^ (1 << j))]);
            #pragma unroll
            for (int j = 5; j < 10; ++j)
                if ((k >> j) & 1)
                    m = fmaxf(m, tokw[j * TP1 + i] + A[(size_t)i * K_ + (k ^ (1 << j))]);
            float s = __expf(prev - m);
            #pragma unroll
            for (int j = 0; j < 5; ++j)
                if ((k >> j) & 1) s += __expf(sRowC[j * K_ + (k ^ (1 << j))] - m);
            #pragma unroll
            for (int j = 5; j < 10; ++j)
                if ((k >> j) & 1)
                    s += __expf(tokw[j * TP1 + i] + A[(size_t)i * K_ + (k ^ (1 << j))] - m);
            const float nv = m + __logf(s);
            A[(size_t)(i + 1) * K_ + k] = nv;
            if (i == T_ && k == K_ - 1) out[0] = nv;   // alpha
        }
        __syncthreads();
    }
}

// ---------------------------------------------------------------------------
// Backward scan (block 1): B[i][k] = LSE( B[i+1][k],
//   for clear bit j<5: row_c[j, k|2^j],  for clear bit j>=5: tok[j,i]+B[i+1][k|2^j] )
// with row_c[j,k] = LSE_d( sb[j,i,d] + B[i+1+d][k] ).
// ---------------------------------------------------------------------------
__device__ void bwd_scan(const float* __restrict__ sb, const float* __restrict__ tokw,
                         float* __restrict__ B, float* __restrict__ out,
                         float* sEa, float* sM, float* sN, float* sRowC)
{
    const int k = threadIdx.x;
    B[(size_t)(T_ + 1) * K_ + k] = (k == K_ - 1) ? 0.0f : NEGV;
    __syncthreads();

    for (int i = T_; i >= 0; --i) {
        const int rem  = T_ - i;
        const int maxD = (rem < L_) ? rem : L_;
        if (k < 16) {
            float M = NEGV;
            if (k < 5) {
                float tmp[LP1];
                #pragma unroll
                for (int d = 0; d < LP1; ++d) {
                    float v = sb[(k * TP1 + i) * LP1 + d];
                    tmp[d] = v; M = fmaxf(M, v);
                }
                #pragma unroll
                for (int d = 0; d < LP1; ++d) sEa[k * 20 + d] = __expf(tmp[d] - M);
                sEa[k * 20 + 17] = 0.f; sEa[k * 20 + 18] = 0.f; sEa[k * 20 + 19] = 0.f;
            } else {
                #pragma unroll
                for (int d = 0; d < 20; ++d) sEa[k * 20 + d] = 0.f;
            }
            sM[k] = M;
        }
        {
            float m = NEGV;
            #pragma unroll
            for (int d = 0; d < LP1; ++d) {
                float v = (d <= maxD) ? B[(size_t)(i + 1 + d) * K_ + k] : NEGV;
                m = fmaxf(m, v);
            }
            sN[k] = m;
        }
        __syncthreads();

        rowc_wmma(k, B, i + 1, +1, maxD, sEa, sM, sN, sRowC);
        __syncthreads();

        {
            const float base = B[(size_t)(i + 1) * K_ + k];
            float m = base;
            #pragma unroll
            for (int j = 0; j < 5; ++j)
                if (!((k >> j) & 1)) m = fmaxf(m, sRowC[j * K_ + (k | (1 << j))]);
            #pragma unroll
            for (int j = 5; j < 10; ++j)
                if (!((k >> j) & 1))
                    m = fmaxf(m, tokw[j * TP1 + i] + B[(size_t)(i + 1) * K_ + (k | (1 << j))]);
            float s = __expf(base - m);
            #pragma unroll
            for (int j = 0; j < 5; ++j)
                if (!((k >> j) & 1)) s += __expf(sRowC[j * K_ + (k | (1 << j))] - m);
            #pragma unroll
            for (int j = 5; j < 10; ++j)
                if (!((k >> j) & 1))
                    s += __expf(tokw[j * TP1 + i] + B[(size_t)(i + 1) * K_ + (k | (1 << j))] - m);
            const float nv = m + __logf(s);
            B[(size_t)i * K_ + k] = nv;
            if (i == 0 && k == 0) out[1] = nv;   // beta
        }
        __syncthreads();
    }
}

__global__ __launch_bounds__(1024)
void scan_kernel(const float* __restrict__ sb, const float* __restrict__ tokw,
                 float* __restrict__ A, float* __restrict__ B, float* __restrict__ out)
{
    __shared__ float sEa[16 * 20];
    __shared__ float sM[16];
    __shared__ float sN[K_];
    __shared__ float sRowC[5 * K_];
    if (blockIdx.x == 0) fwd_scan(sb, tokw, A, out, sEa, sM, sN, sRowC);
    else                 bwd_scan(sb, tokw, B, out, sEa, sM, sN, sRowC);
}

// ---------------------------------------------------------------------------
// Marginals: block per (t, j). pair[d] = LSE over 512 src states (bit j clear)
// of A[t][s] + B[min(t+d+1, T+1)][s | 2^j]; online-LSE + LDS tree reduction.
//   marg_row[j][t][d] = sb[j][t][d] + pair[d] - alpha
//   marg_col[j][t]    = tok_w[j][t] + pair[0] - alpha
// ---------------------------------------------------------------------------
__global__ __launch_bounds__(256)
void marg_kernel(const float* __restrict__ sb, const float* __restrict__ tokw,
                 const float* __restrict__ A, const float* __restrict__ B,
                 float* __restrict__ out)
{
    __shared__ float sm[256 * LP1];
    __shared__ float ss[256 * LP1];
    const int t   = blockIdx.x;   // 0..T
    const int j   = blockIdx.y;   // 0..S-1
    const int tid = threadIdx.x;

    float m[LP1], s[LP1];
    #pragma unroll
    for (int d = 0; d < LP1; ++d) { m[d] = -3.0e38f; s[d] = 0.0f; }

    #pragma unroll
    for (int it = 0; it < 2; ++it) {
        const int n     = tid + it * 256;                 // 0..511
        const int sfull = ((n >> j) << (j + 1)) | (n & ((1 << j) - 1));
        const int tgt   = sfull | (1 << j);
        const float a   = A[(size_t)t * K_ + sfull];
        #pragma unroll
        for (int d = 0; d < LP1; ++d) {
            int trow = t + d + 1; if (trow > T_ + 1) trow = T_ + 1;
            const float v = a + B[(size_t)trow * K_ + tgt];
            if (v > m[d]) { s[d] = s[d] * __expf(m[d] - v) + 1.0f; m[d] = v; }
            else          { s[d] += __expf(v - m[d]); }
        }
    }

    #pragma unroll
    for (int d = 0; d < LP1; ++d) { sm[tid * LP1 + d] = m[d]; ss[tid * LP1 + d] = s[d]; }
    __syncthreads();

    for (int str = 128; str > 0; str >>= 1) {
        if (tid < str) {
            #pragma unroll
            for (int d = 0; d < LP1; ++d) {
                float m1 = sm[tid * LP1 + d],        s1 = ss[tid * LP1 + d];
                float m2 = sm[(tid + str) * LP1 + d], s2 = ss[(tid + str) * LP1 + d];
                if (m2 > m1) { float tm = m1; m1 = m2; m2 = tm; float ts = s1; s1 = s2; s2 = ts; }
                sm[tid * LP1 + d] = m1;
                ss[tid * LP1 + d] = s1 + s2 * __expf(m2 - m1);
            }
        }
        __syncthreads();
    }

    const float alpha = out[0];
    if (tid < LP1) {
        const float pair = sm[tid] + __logf(ss[tid]);
        out[2 + ((size_t)j * TP1 + t) * LP1 + tid] =
            sb[((size_t)j * TP1 + t) * LP1 + tid] + pair - alpha;
        if (tid == 0)
            out[2 + (size_t)S_ * TP1 * LP1 + (size_t)j * TP1 + t] =
                tokw[(size_t)j * TP1 + t] + pair - alpha;
    }
}

// ---------------------------------------------------------------------------
extern "C" void kernel_launch(void* const* d_in, const int* in_sizes, int n_in,
                              void* d_out, int out_size, void* d_ws, size_t ws_size,
                              hipStream_t stream)
{
    (void)in_sizes; (void)n_in; (void)out_size; (void)ws_size;
    const float* sb   = (const float*)d_in[0];   // (10, 1025, 17)
    const float* tokw = (const float*)d_in[1];   // (10, 1025)
    float* out = (float*)d_out;                  // [alpha, beta, marg_row, marg_col]
    float* A   = (float*)d_ws;                   // (1026, 1024)
    float* B   = A + (size_t)ROWS * K_;          // (1026, 1024)

    // Forward and backward scans run concurrently as two independent blocks.
    scan_kernel<<<2, K_, 0, stream>>>(sb, tokw, A, B, out);
    // Marginals: one block per (t, slot).
    dim3 grid(TP1, S_);
    marg_kernel<<<grid, 256, 0, stream>>>(sb, tokw, A, B, out);
}